// ChebyUpsample_32255204393781
// MI455X (gfx1250) — compile-verified
//
#include <hip/hip_runtime.h>
#include <math.h>

typedef float v2f __attribute__((ext_vector_type(2)));
typedef float v4f __attribute__((ext_vector_type(4)));
typedef float v8f __attribute__((ext_vector_type(8)));

#define T_IN   32768
#define T_OUT  65536
#define NROW   256
#define CHUNK  2048
#define NCHUNK (T_OUT / CHUNK)   /* 32 chunks per row  */
#define WARMG  1                 /* 512 warm-up samples (slowest pole |z|=0.933 -> 3e-16 decay) */
#define MAING  (CHUNK / 512)     /* 4 main groups (512 samples each) per chunk */
#define NG     (WARMG + MAING)   /* 5 groups per wave */
#define NWAVE  (NROW * NCHUNK)   /* 8192 waves */

// ws layout (floats), block size = 32 samples:
//   [   0,1024)  H32 : 32x32 zero-state block response (row-major [m][k])
//   [1024,1536)  T32 : 32x16 state->output (cols 8..15 zero)
//   [1536,2048)  P32 : 16x32 input->state  (rows 8..15 zero)
//   [2048,2112)  F   : 8x8 A^32 (row-major)

__device__ static void cheby_step(double s[8], double x, double* y,
                                  const double* B0, const double* B1, const double* B2,
                                  const double* A1, const double* A2) {
    double in = x;
    for (int sec = 0; sec < 4; ++sec) {
        double yy = B0[sec] * in + s[2 * sec];
        double z1 = B1[sec] * in - A1[sec] * yy + s[2 * sec + 1];
        double z2 = B2[sec] * in - A2[sec] * yy;
        s[2 * sec] = z1;
        s[2 * sec + 1] = z2;
        in = yy;
    }
    *y = in;
}

__global__ void cheby_setup(float* ws) {
    if (threadIdx.x != 0 || blockIdx.x != 0) return;
    const int    N  = 8;
    const double PI = 3.14159265358979323846;
    double rp = 0.05, Wn = 0.5;
    double eps = sqrt(pow(10.0, 0.1 * rp) - 1.0);
    double mu  = asinh(1.0 / eps) / (double)N;

    double pr[8], pim[8];
    for (int i = 0; i < 8; ++i) {
        int    m  = 2 * i - (N - 1);
        double th = PI * (double)m / (2.0 * N);
        pr[i]  = -sinh(mu) * cos(th);
        pim[i] = -cosh(mu) * sin(th);
    }
    double kr = 1.0, ki = 0.0;
    for (int i = 0; i < 8; ++i) {
        double ar = -pr[i], ai = -pim[i];
        double nr = kr * ar - ki * ai, ni = kr * ai + ki * ar;
        kr = nr; ki = ni;
    }
    double k  = kr / sqrt(1.0 + eps * eps);
    double fs = 2.0;
    double warped = 2.0 * fs * tan(PI * Wn / fs);
    for (int i = 0; i < 8; ++i) { pr[i] *= warped; pim[i] *= warped; }
    for (int i = 0; i < N; ++i) k *= warped;
    double fs2 = 2.0 * fs;
    double pzr[8], pzi[8], dr = 1.0, di = 0.0;
    for (int i = 0; i < 8; ++i) {
        double nr2 = fs2 + pr[i], ni2 = pim[i];
        double dr2 = fs2 - pr[i], di2 = -pim[i];
        double den = dr2 * dr2 + di2 * di2;
        pzr[i] = (nr2 * dr2 + ni2 * di2) / den;
        pzi[i] = (ni2 * dr2 - nr2 * di2) / den;
        double tr = dr * dr2 - di * di2, ti = dr * di2 + di * dr2;
        dr = tr; di = ti;
    }
    double kz = k * (dr / (dr * dr + di * di));

    double B0[4], B1[4], B2[4], A1s[4], A2s[4];
    int nsec = 0;
    for (int i = 0; i < 8; ++i) {
        if (pzi[i] > 0.0) {
            B0[nsec] = 1.0; B1[nsec] = 2.0; B2[nsec] = 1.0;
            A1s[nsec] = -2.0 * pzr[i];
            A2s[nsec] = pzr[i] * pzr[i] + pzi[i] * pzi[i];
            ++nsec;
        }
    }
    B0[0] *= kz; B1[0] *= kz; B2[0] *= kz;

    double A[8][8], b[8], c[8], d;
    for (int col = 0; col < 8; ++col) {
        double s[8] = {0, 0, 0, 0, 0, 0, 0, 0};
        s[col] = 1.0;
        double y;
        cheby_step(s, 0.0, &y, B0, B1, B2, A1s, A2s);
        for (int i = 0; i < 8; ++i) A[i][col] = s[i];
        c[col] = y;
    }
    {
        double s[8] = {0, 0, 0, 0, 0, 0, 0, 0};
        double y;
        cheby_step(s, 1.0, &y, B0, B1, B2, A1s, A2s);
        for (int i = 0; i < 8; ++i) b[i] = s[i];
        d = y;
    }

    // Trow[m] = c^T A^m  (m = 0..31)
    double Trow[32][8];
    for (int i = 0; i < 8; ++i) Trow[0][i] = c[i];
    for (int m = 1; m < 32; ++m)
        for (int kk = 0; kk < 8; ++kk) {
            double acc = 0.0;
            for (int i = 0; i < 8; ++i) acc += Trow[m - 1][i] * A[i][kk];
            Trow[m][kk] = acc;
        }
    // impulse response g[n] = c^T A^n b
    double g[32];
    for (int n = 0; n < 32; ++n) {
        double acc = 0.0;
        for (int i = 0; i < 8; ++i) acc += Trow[n][i] * b[i];
        g[n] = acc;
    }
    // H32 (32x32)
    for (int t = 0; t < 32; ++t)
        for (int kk = 0; kk < 32; ++kk) {
            double h = (t < kk) ? 0.0 : ((t == kk) ? d : g[t - kk - 1]);
            ws[t * 32 + kk] = (float)h;
        }
    // T32 (32x16, state cols padded)
    for (int m = 0; m < 32; ++m)
        for (int i = 0; i < 16; ++i)
            ws[1024 + m * 16 + i] = (i < 8) ? (float)Trow[m][i] : 0.0f;
    // P32 (16x32): P[:,k] = A^(31-k) b
    double w[32][8];
    for (int i = 0; i < 8; ++i) w[0][i] = b[i];
    for (int n = 1; n < 32; ++n)
        for (int i = 0; i < 8; ++i) {
            double acc = 0.0;
            for (int kk = 0; kk < 8; ++kk) acc += A[i][kk] * w[n - 1][kk];
            w[n][i] = acc;
        }
    for (int i = 0; i < 16; ++i)
        for (int kk = 0; kk < 32; ++kk)
            ws[1536 + i * 32 + kk] = (i < 8) ? (float)w[31 - kk][i] : 0.0f;
    // F = A^32 (square 5 times)
    double M[8][8], M2[8][8];
    for (int i = 0; i < 8; ++i)
        for (int kk = 0; kk < 8; ++kk) M[i][kk] = A[i][kk];
    for (int r = 0; r < 5; ++r) {
        for (int i = 0; i < 8; ++i)
            for (int kk = 0; kk < 8; ++kk) {
                double acc = 0.0;
                for (int q = 0; q < 8; ++q) acc += M[i][q] * M[q][kk];
                M2[i][kk] = acc;
            }
        for (int i = 0; i < 8; ++i)
            for (int kk = 0; kk < 8; ++kk) M[i][kk] = M2[i][kk];
    }
    for (int i = 0; i < 8; ++i)
        for (int kk = 0; kk < 8; ++kk) ws[2048 + i * 8 + kk] = (float)M[i][kk];
}

__global__ __launch_bounds__(256) void cheby_filter(const float* __restrict__ x,
                                                    float* __restrict__ out,
                                                    const float* __restrict__ ws) {
    // per-wave LDS staging: input double-buffered (2 x 256 floats), output 512 floats
    __shared__ float s_in[8][2][256];
    __shared__ float s_out[8][512];

    const int lane = threadIdx.x & 31;
    const int w    = threadIdx.x >> 5;
    const int wid  = (int)((blockIdx.x * blockDim.x + threadIdx.x) >> 5);
    const int row   = wid / NCHUNK;
    const int chunk = wid % NCHUNK;
    const float* xr = x + (size_t)row * T_IN;
    float* outr     = out + (size_t)row * T_OUT;

    const int hi = lane >> 4;
    const int lj = lane & 15;
    const int ko = 4 * 0 + hi * 2; (void)ko;

    // A-operands: lanes 0-15 = rows, .x/.y = K,K+1 (lanes 16-31: K+2,K+3)
    v2f aH[2][8], aT[2][4], aP[8];
#pragma unroll
    for (int t = 0; t < 2; ++t)
#pragma unroll
        for (int gq = 0; gq < 8; ++gq) {
            int idx = (16 * t + lj) * 32 + 4 * gq + hi * 2;
            aH[t][gq].x = ws[idx];
            aH[t][gq].y = ws[idx + 1];
        }
#pragma unroll
    for (int t = 0; t < 2; ++t)
#pragma unroll
        for (int gq = 0; gq < 4; ++gq) {
            int idx = 1024 + (16 * t + lj) * 16 + 4 * gq + hi * 2;
            aT[t][gq].x = ws[idx];
            aT[t][gq].y = ws[idx + 1];
        }
#pragma unroll
    for (int gq = 0; gq < 8; ++gq) {
        int idx = 1536 + lj * 32 + 4 * gq + hi * 2;
        aP[gq].x = ws[idx];
        aP[gq].y = ws[idx + 1];
    }
    // F (8x8, wave-uniform) pinned into SGPRs to keep VGPR pressure low
    int Fs[64];
#pragma unroll
    for (int i = 0; i < 64; ++i)
        Fs[i] = __builtin_amdgcn_readfirstlane(__float_as_int(ws[2048 + i]));

    float s[8];
#pragma unroll
    for (int i = 0; i < 8; ++i) s[i] = 0.0f;

    const int tau_begin = chunk * CHUNK - WARMG * 512;
    const unsigned in_base = (unsigned)(unsigned long long)&s_in[w][0][0];

    // async load of one group's 256 consecutive input floats (2 x coalesced 512B)
    auto issue_in = [&](int gi, int buf) {
        int tau0 = tau_begin + gi * 512;
        int gb   = (T_OUT - 512 - tau0) >> 1;      // 256-float aligned
        if (gb > T_IN - 256) gb = T_IN - 256;      // clamp (chunk-0 warm-up; values zeroed at use)
        const float* gp0 = xr + gb + lane * 4;
        const float* gp1 = xr + gb + 128 + lane * 4;
        unsigned l0 = in_base + (unsigned)buf * 1024u + (unsigned)lane * 16u;
        unsigned l1 = l0 + 512u;
        asm volatile("global_load_async_to_lds_b128 %0, %1, off"
                     :: "v"(l0), "v"(gp0) : "memory");
        asm volatile("global_load_async_to_lds_b128 %0, %1, off"
                     :: "v"(l1), "v"(gp1) : "memory");
    };

    issue_in(0, 0);

    for (int gi = 0; gi < NG; ++gi) {
        const int  cur     = gi & 1;
        const int  tau0    = tau_begin + gi * 512;
        const bool mainreg = (gi >= WARMG);

        // drain this group's async loads
        asm volatile("s_wait_asynccnt 0x0" ::: "memory");

        // X[k][j] (32x16) from LDS; repeat-2 upsample => k,k+1 share one input sample.
        // LDS float offset = (511 - (32 j + k)) >> 1, group-invariant.
        v2f xb[8];
#pragma unroll
        for (int gq = 0; gq < 8; ++gq) {
            int off  = (511 - (lj * 32 + 4 * gq + hi * 2)) >> 1;
            float xv = s_in[w][cur][off];
            if (tau0 < 0) xv = 0.0f;               // wave-uniform zero for warm-up underrun
            xb[gq].x = xv;
            xb[gq].y = xv;
        }

        // prefetch next group into the other buffer (overlaps with compute below)
        if (gi + 1 < NG) issue_in(gi + 1, cur ^ 1);

        // U = P32 * X : lane j holds u_j[0..7]
        v8f uacc;
#pragma unroll
        for (int i = 0; i < 8; ++i) uacc[i] = 0.0f;
#pragma unroll
        for (int gq = 0; gq < 8; ++gq)
            uacc = __builtin_amdgcn_wmma_f32_16x16x4_f32(false, aP[gq], false, xb[gq],
                                                         (short)0, uacc, false, false);

        v8f yaccA, yaccB;
#pragma unroll
        for (int i = 0; i < 8; ++i) { yaccA[i] = 0.0f; yaccB[i] = 0.0f; }
        if (mainreg) {
#pragma unroll
            for (int gq = 0; gq < 8; ++gq)
                yaccA = __builtin_amdgcn_wmma_f32_16x16x4_f32(false, aH[0][gq], false, xb[gq],
                                                              (short)0, yaccA, false, false);
#pragma unroll
            for (int gq = 0; gq < 8; ++gq)
                yaccB = __builtin_amdgcn_wmma_f32_16x16x4_f32(false, aH[1][gq], false, xb[gq],
                                                              (short)0, yaccB, false, false);
        }

        // serial 16-step block scan: s_{j+1} = F s_j + u_j ; capture S (B-layout) for T*S
        v2f sb[4];
#pragma unroll
        for (int gq = 0; gq < 4; ++gq) { sb[gq].x = 0.0f; sb[gq].y = 0.0f; }

        for (int j = 0; j < 16; ++j) {
            if (mainreg && (lj == j)) {
                sb[0].x = hi ? s[2] : s[0];
                sb[0].y = hi ? s[3] : s[1];
                sb[1].x = hi ? s[6] : s[4];
                sb[1].y = hi ? s[7] : s[5];
            }
            float uu[8];
#pragma unroll
            for (int i = 0; i < 8; ++i)
                uu[i] = __int_as_float(
                    __builtin_amdgcn_readlane(__float_as_int(uacc[i]), j));
            float ns[8];
#pragma unroll
            for (int i = 0; i < 8; ++i) {
                float acc = uu[i];
#pragma unroll
                for (int kk = 0; kk < 8; ++kk)
                    acc = fmaf(__int_as_float(Fs[i * 8 + kk]), s[kk], acc);
                ns[i] = acc;
            }
#pragma unroll
            for (int i = 0; i < 8; ++i) s[i] = ns[i];
        }

        if (mainreg) {
            // Y += T32 * S (both row tiles)
#pragma unroll
            for (int gq = 0; gq < 4; ++gq)
                yaccA = __builtin_amdgcn_wmma_f32_16x16x4_f32(false, aT[0][gq], false, sb[gq],
                                                              (short)0, yaccA, false, false);
#pragma unroll
            for (int gq = 0; gq < 4; ++gq)
                yaccB = __builtin_amdgcn_wmma_f32_16x16x4_f32(false, aT[1][gq], false, sb[gq],
                                                              (short)0, yaccB, false, false);

            // scatter D-layout into per-wave LDS tile (time-reversed, pairs adjacent)
            const int LA = lj * 32 + hi * 8;        // tile A: m = v + hi*8
            const int LB = LA + 16;                 // tile B: m = 16 + v + hi*8
#pragma unroll
            for (int v = 0; v < 8; v += 2) {
                v2f pa, pb;
                pa.x = yaccA[v + 1]; pa.y = yaccA[v];
                pb.x = yaccB[v + 1]; pb.y = yaccB[v];
                *(v2f*)&s_out[w][511 - LA - (v + 1)] = pa;
                *(v2f*)&s_out[w][511 - LB - (v + 1)] = pb;
            }
            // coalesced writeback: 4 x 512B stores (DS in-order makes the roundtrip safe)
            const int gbo = T_OUT - 512 - tau0;
#pragma unroll
            for (int r = 0; r < 4; ++r) {
                v4f dv = *(v4f*)&s_out[w][r * 128 + lane * 4];
                *(v4f*)&outr[gbo + r * 128 + lane * 4] = dv;
            }
        }
    }
}

extern "C" void kernel_launch(void* const* d_in, const int* in_sizes, int n_in,
                              void* d_out, int out_size, void* d_ws, size_t ws_size,
                              hipStream_t stream) {
    const float* x  = (const float*)d_in[0];
    float*       o  = (float*)d_out;
    float*       ws = (float*)d_ws;

    cheby_setup<<<1, 1, 0, stream>>>(ws);

    const int threads = 256;
    const int blocks  = (NWAVE * 32) / threads;   // 1024
    cheby_filter<<<blocks, threads, 0, stream>>>(x, o, ws);
}